// SkipGram_40948218200523
// MI455X (gfx1250) — compile-verified
//
#include <hip/hip_runtime.h>
#include <math.h>

typedef __attribute__((ext_vector_type(2))) float v2f;
typedef __attribute__((ext_vector_type(8))) float v8f;

#define EMBED   128
#define NUM_NEG 64
#define S_VALID 65           // 1 target + 64 negatives
#define S_PAD   80           // 5 WMMA tiles of 16
#define WPB     8            // waves per block (wave32)

// One wave32 per batch row.
// Logits via V_WMMA_F32_16X16X4_F32 with A = embed broadcast to all 16 rows:
//   A[m,k] = embed[kbase+k]  ->  D[m,n] = dot(embed_chunk, w[sample_n]_chunk) for every m.
// A layout (ISA 7.12.2, 32-bit A 16x4): VGPR0 = K=(half?2:0), VGPR1 = K+1  -> contiguous float2.
// B layout (4x16, mirrored): lanes hold N=lane&15, VGPR0 = K=(half?2:0), VGPR1 = K+1 -> contiguous float2.
__global__ __launch_bounds__(WPB * 32) void sg_softmax_wmma(
    const int*   __restrict__ target,
    const int*   __restrict__ context,
    const int*   __restrict__ negatives,
    const float* __restrict__ emb_table,
    const float* __restrict__ w_table,
    const float* __restrict__ b_table,
    float*       __restrict__ out,
    int batch)
{
    __shared__ float lds_e[WPB][EMBED];   // staged context embedding per wave
    __shared__ float lds_l[WPB][S_PAD];   // logits per wave

    const int lane = threadIdx.x & 31;
    const int wv   = threadIdx.x >> 5;
    const int b    = blockIdx.x * WPB + wv;
    if (b >= batch) return;               // uniform per wave -> EXEC stays all-ones for WMMA

    const int half = lane >> 4;           // which 16-lane half
    const int ln   = lane & 15;

    // ---- stage context embedding row into LDS (one float4 per lane, 512B burst) ----
    const int ctx = context[b];
    ((float4*)lds_e[wv])[lane] = ((const float4*)(emb_table + (size_t)ctx * EMBED))[lane];

    // ---- per-tile sample indices; pads alias the target row (already hot in L0) ----
    const int tgt = target[b];
    const float* wrow[5];
#pragma unroll
    for (int j = 0; j < 5; ++j) {
        int s   = j * 16 + ln;
        int idx = (s == 0) ? tgt
                : (s <= NUM_NEG ? negatives[(size_t)b * NUM_NEG + (s - 1)] : tgt);
        wrow[j] = w_table + (size_t)idx * EMBED;
    }

    asm volatile("s_wait_dscnt 0" ::: "memory");   // embed staged -> safe to ds_load

    v8f c[5];
#pragma unroll
    for (int j = 0; j < 5; ++j) c[j] = v8f{};

    const int cb = 2 * half;              // column base within each K=4 chunk
#pragma unroll 4
    for (int t = 0; t < EMBED / 4; ++t) {
        v2f a = *(const v2f*)&lds_e[wv][4 * t + cb];
#pragma unroll
        for (int j = 0; j < 5; ++j) {
            v2f bm = *(const v2f*)(wrow[j] + 4 * t + cb);   // 256B burst per instruction
            c[j] = __builtin_amdgcn_wmma_f32_16x16x4_f32(
                       false, a, false, bm, (short)0, c[j], false, false);
        }
    }

    // D rows are identical; row M=0 sits in VGPR0 of lanes 0..15 (N = lane)
    if (lane < 16) {
#pragma unroll
        for (int j = 0; j < 5; ++j) lds_l[wv][j * 16 + lane] = c[j][0];
    }
    asm volatile("s_wait_dscnt 0" ::: "memory");

    // ---- bias + softmax over 65 valid entries (pads masked to -inf) ----
    // lane covers s = lane, lane+32, and (lane==0) s = 64
    const int s0 = lane, s1 = lane + 32;
    const int i0 = (s0 == 0) ? tgt : negatives[(size_t)b * NUM_NEG + (s0 - 1)];
    const int i1 = negatives[(size_t)b * NUM_NEG + (s1 - 1)];
    float x0 = lds_l[wv][s0] + b_table[i0];
    float x1 = lds_l[wv][s1] + b_table[i1];
    float x2 = -INFINITY;
    if (lane == 0) x2 = lds_l[wv][64] + b_table[negatives[(size_t)b * NUM_NEG + 63]];

    float m = fmaxf(fmaxf(x0, x1), x2);
#pragma unroll
    for (int off = 16; off > 0; off >>= 1) m = fmaxf(m, __shfl_xor(m, off, 32));

    float e0 = __expf(x0 - m);
    float e1 = __expf(x1 - m);
    float e2 = (lane == 0) ? __expf(x2 - m) : 0.0f;
    float ssum = e0 + e1 + e2;
#pragma unroll
    for (int off = 16; off > 0; off >>= 1) ssum += __shfl_xor(ssum, off, 32);
    const float inv = 1.0f / ssum;

    float* orow = out + (size_t)b * S_VALID;
    orow[s0] = e0 * inv;
    orow[s1] = e1 * inv;
    if (lane == 0) orow[64] = e2 * inv;
}

extern "C" void kernel_launch(void* const* d_in, const int* in_sizes, int n_in,
                              void* d_out, int out_size, void* d_ws, size_t ws_size,
                              hipStream_t stream)
{
    (void)n_in; (void)out_size; (void)d_ws; (void)ws_size;
    const int*   target    = (const int*)  d_in[0];
    const int*   context   = (const int*)  d_in[1];
    const int*   negatives = (const int*)  d_in[2];
    const float* emb_table = (const float*)d_in[3];
    const float* w_table   = (const float*)d_in[4];
    const float* b_table   = (const float*)d_in[5];
    float*       out       = (float*)      d_out;

    const int batch  = in_sizes[0];                 // 16384
    const int blocks = (batch + WPB - 1) / WPB;     // one wave32 per row
    sg_softmax_wmma<<<blocks, WPB * 32, 0, stream>>>(
        target, context, negatives, emb_table, w_table, b_table, out, batch);
}